// ThermometerEncodingPytorch_76785425318126
// MI455X (gfx1250) — compile-verified
//
#include <hip/hip_runtime.h>

// Thermometer encoding, MI455X (gfx1250).
// x: (32, 3, 512, 512) f32 in [0,1)  ->  out: (32, 30, 512, 512) f32
// out[b, c*10 + k, h, w] = (k == 0) ? 1.0f : (x[b,c,h,w] > k/10 ? 1.0f : 0.0f)
//
// Pure streaming expansion: ~0.1 GB in, ~1.0 GB out => HBM-store bound
// (~48 us floor at 23.3 TB/s). Strategy: b128 load + 10x b128 stores per
// thread, all non-temporal (write-once data, don't pollute the 192 MB L2),
// fully coalesced within each of the 10 destination planes.

typedef float v4f __attribute__((ext_vector_type(4)));

#define NS        10
#define HW        (512 * 512)   // 262144 = 1 << 18
#define HW_SHIFT  18

__global__ __launch_bounds__(256, 1)
void thermo10_b128_nt(const float* __restrict__ x,
                      float* __restrict__ out,
                      unsigned int n4 /* number of float4 input elements */) {
    const unsigned int g = blockIdx.x * blockDim.x + threadIdx.x;
    if (g >= n4) return;

    const size_t i     = (size_t)g << 2;          // input float index (4 per thread)
    const size_t plane = i >> HW_SHIFT;           // b*3 + c   (0..95)
    const size_t p     = i & (size_t)(HW - 1);    // pixel within plane

    // Read-once input: non-temporal b128 load.
    const v4f v = __builtin_nontemporal_load((const v4f*)(x + i));

    // out base for this (b, c): channel (c*10 + k) plane k is HW floats apart.
    float* ob = out + plane * (size_t)(NS * HW) + p;

    // k = 0 plane is identically 1.0 (pos >= 0 always true).
    const v4f ones = {1.0f, 1.0f, 1.0f, 1.0f};
    __builtin_nontemporal_store(ones, (v4f*)ob);

#pragma unroll
    for (int k = 1; k < NS; ++k) {
        // Match JAX: threshold = float(k) / 10.0f (constant-folded, RNE).
        const float t = (float)k / 10.0f;
        v4f r;
        r.x = (v.x > t) ? 1.0f : 0.0f;
        r.y = (v.y > t) ? 1.0f : 0.0f;
        r.z = (v.z > t) ? 1.0f : 0.0f;
        r.w = (v.w > t) ? 1.0f : 0.0f;
        __builtin_nontemporal_store(r, (v4f*)(ob + (size_t)k * HW));
    }
}

extern "C" void kernel_launch(void* const* d_in, const int* in_sizes, int n_in,
                              void* d_out, int out_size, void* d_ws, size_t ws_size,
                              hipStream_t stream) {
    (void)n_in; (void)out_size; (void)d_ws; (void)ws_size;

    const float* x   = (const float*)d_in[0];
    float*       out = (float*)d_out;

    const unsigned int n  = (unsigned int)in_sizes[0];  // 32*3*512*512, divisible by 4
    const unsigned int n4 = n >> 2;                     // float4 elements
    const unsigned int threads = 256;                   // 8 wave32 waves per block
    const unsigned int blocks  = (n4 + threads - 1) / threads;

    thermo10_b128_nt<<<blocks, threads, 0, stream>>>(x, out, n4);
}